// NGCF_30502857736677
// MI455X (gfx1250) — compile-verified
//
#include <hip/hip_runtime.h>

#define USER_NUM 52643
#define ITEM_NUM 91599
#define NNODES   (USER_NUM + ITEM_NUM)   // 144242
#define DFEAT    64

typedef float v2f __attribute__((ext_vector_type(2)));
typedef float v8f __attribute__((ext_vector_type(8)));

// ---------------------------------------------------------------------------
// embed0 = concat(embed_user, embed_item)
// ---------------------------------------------------------------------------
__global__ void concat_embed_kernel(const float* __restrict__ eu,
                                    const float* __restrict__ ei,
                                    float* __restrict__ e0) {
  size_t i  = (size_t)blockIdx.x * blockDim.x + threadIdx.x;
  size_t nu = (size_t)USER_NUM * DFEAT;
  size_t nt = (size_t)NNODES * DFEAT;
  if (i < nu)      e0[i] = eu[i];
  else if (i < nt) e0[i] = ei[i - nu];
}

// P1 = src (identity term of L+I), P2 = 0; atomics accumulate on top.
__global__ void init_p_kernel(const float* __restrict__ src,
                              float* __restrict__ p1, float* __restrict__ p2) {
  size_t i = (size_t)blockIdx.x * blockDim.x + threadIdx.x;
  if (i < (size_t)NNODES * DFEAT) { p1[i] = src[i]; p2[i] = 0.0f; }
}

__global__ void zero_loss_kernel(float* loss) {
  if (threadIdx.x == 0 && blockIdx.x == 0) *loss = 0.0f;
}

// ---------------------------------------------------------------------------
// Dual SpMM: P1 += L@X, P2 += L@(X*X). One wave per edge, float2 per lane.
// X (36.9 MB) is L2-resident on MI455X (192 MB L2), so the col-gather hits L2
// and the scatter is global_atomic_add_f32.
// ---------------------------------------------------------------------------
__global__ void spmm_dual_kernel(const int* __restrict__ rows,
                                 const int* __restrict__ cols,
                                 const float* __restrict__ vals,
                                 const float* __restrict__ X,
                                 float* __restrict__ P1, float* __restrict__ P2,
                                 int E) {
  int wave = (int)(((size_t)blockIdx.x * blockDim.x + threadIdx.x) >> 5);
  int lane = threadIdx.x & 31;
  if (wave >= E) return;                       // wave-uniform branch
  int   r = rows[wave];
  int   c = cols[wave];
  float v = vals[wave];
  const float* xp = X + (size_t)c * DFEAT + lane * 2;
  float x0 = xp[0], x1 = xp[1];
  float* p1 = P1 + (size_t)r * DFEAT + lane * 2;
  float* p2 = P2 + (size_t)r * DFEAT + lane * 2;
  unsafeAtomicAdd(p1,     v * x0);
  unsafeAtomicAdd(p1 + 1, v * x1);
  unsafeAtomicAdd(p2,     v * x0 * x0);
  unsafeAtomicAdd(p2 + 1, v * x1 * x1);
}

// ---------------------------------------------------------------------------
// Fused dense layer: Out = relu(P1 @ W^T + b + P2 @ Wi^T + bi)
// One wave computes one 16x16 output tile via V_WMMA_F32_16X16X4_F32.
// K = 64 -> 16 k-steps, two weight matrices -> 32 chained WMMAs per wave.
//
// f32 A-frag layout (16x4): lanes 0-15 carry M=lane, K={k0,k0+1};
//                           lanes 16-31 carry M=lane-16, K={k0+2,k0+3}.
// f32 B-frag layout (4x16): lanes 0-15 carry N=lane, K={k0,k0+1};
//                           lanes 16-31 carry N=lane-16, K={k0+2,k0+3}.
// Out needs B[k][j] = W[j][k], so the B-frag is a contiguous float2 from row j
// of the row-major W. D-frag: c[r] -> row = row0 + r + 8*(lane>>4), col = lane&15.
// ---------------------------------------------------------------------------
__global__ void gemm_bias_relu_kernel(const float* __restrict__ P1,
                                      const float* __restrict__ P2,
                                      const float* __restrict__ W,
                                      const float* __restrict__ b,
                                      const float* __restrict__ Wi,
                                      const float* __restrict__ bi,
                                      float* __restrict__ Out, int Nrows) {
  int wave = (int)(((size_t)blockIdx.x * blockDim.x + threadIdx.x) >> 5);
  int lane = threadIdx.x & 31;
  int numRowTiles = (Nrows + 15) >> 4;
  int rowTile = wave >> 2;                     // 4 col-tiles (64/16) per row-tile
  int colTile = wave & 3;
  if (rowTile >= numRowTiles) return;          // wave-uniform; EXEC stays full
  int row0 = rowTile << 4;
  int col0 = colTile << 4;
  int lmod = lane & 15;
  int lhi  = lane >> 4;                        // 0 or 1

  int arow = row0 + lmod;                      // clamp ragged tail (loads only)
  if (arow >= Nrows) arow = Nrows - 1;
  int bcol = col0 + lmod;                      // always < 64

  const float* a1base = P1 + (size_t)arow * DFEAT + lhi * 2;
  const float* a2base = P2 + (size_t)arow * DFEAT + lhi * 2;
  const float* wbase  = W  + (size_t)bcol * DFEAT + lhi * 2;
  const float* wibase = Wi + (size_t)bcol * DFEAT + lhi * 2;

  v8f c = {};
#pragma unroll
  for (int kk = 0; kk < 16; ++kk) {
    int k0 = kk * 4;
    v2f a1 = *(const v2f*)(a1base + k0);
    v2f bw = *(const v2f*)(wbase  + k0);
    c = __builtin_amdgcn_wmma_f32_16x16x4_f32(false, a1, false, bw,
                                              (short)0, c, false, false);
    v2f a2  = *(const v2f*)(a2base + k0);
    v2f bwi = *(const v2f*)(wibase + k0);
    c = __builtin_amdgcn_wmma_f32_16x16x4_f32(false, a2, false, bwi,
                                              (short)0, c, false, false);
  }

  float bias = b[bcol] + bi[bcol];             // bias depends only on out-col
#pragma unroll
  for (int r = 0; r < 8; ++r) {
    int row = row0 + r + 8 * lhi;
    if (row < Nrows) {
      float v = c[r] + bias;
      Out[(size_t)row * DFEAT + col0 + lmod] = v > 0.0f ? v : 0.0f;
    }
  }
}

// ---------------------------------------------------------------------------
// Final gather + dot + loss. One wave per batch element; each lane owns 2 of
// the 64 dims in each of the 3 concatenated segments (embed0, g1, g2).
// ---------------------------------------------------------------------------
__global__ void pred_loss_kernel(const int* __restrict__ user,
                                 const int* __restrict__ item_i,
                                 const int* __restrict__ item_j,
                                 const float* __restrict__ E0,
                                 const float* __restrict__ G1,
                                 const float* __restrict__ G2,
                                 float* __restrict__ out, int batch) {
  int wave = (int)(((size_t)blockIdx.x * blockDim.x + threadIdx.x) >> 5);
  int lane = threadIdx.x & 31;
  if (wave >= batch) return;
  int u  = user[wave];
  int ii = USER_NUM + item_i[wave];
  int jj = USER_NUM + item_j[wave];

  const float* mats[3] = {E0, G1, G2};
  float di = 0.0f, dj = 0.0f;
#pragma unroll
  for (int s = 0; s < 3; ++s) {
    const float* m = mats[s];
    float2 ue = *(const float2*)(m + (size_t)u  * DFEAT + lane * 2);
    float2 ie = *(const float2*)(m + (size_t)ii * DFEAT + lane * 2);
    float2 je = *(const float2*)(m + (size_t)jj * DFEAT + lane * 2);
    di += ue.x * ie.x + ue.y * ie.y;
    dj += ue.x * je.x + ue.y * je.y;
  }
#pragma unroll
  for (int off = 16; off >= 1; off >>= 1) {
    di += __shfl_down(di, off, 32);
    dj += __shfl_down(dj, off, 32);
  }
  if (lane == 0) {
    out[wave]         = di;
    out[batch + wave] = dj;
    float y  = dj - di;                         // -(pred_i - pred_j)
    float sp = fmaxf(y, 0.0f) + log1pf(expf(-fabsf(y)));  // softplus(-x)
    atomicAdd(&out[2 * batch], sp);
  }
}

// ---------------------------------------------------------------------------
extern "C" void kernel_launch(void* const* d_in, const int* in_sizes, int n_in,
                              void* d_out, int out_size, void* d_ws, size_t ws_size,
                              hipStream_t stream) {
  const int*   user   = (const int*)d_in[0];
  const int*   item_i = (const int*)d_in[1];
  const int*   item_j = (const int*)d_in[2];
  const int*   rows   = (const int*)d_in[3];
  const int*   cols   = (const int*)d_in[4];
  const float* vals   = (const float*)d_in[5];
  const float* eu     = (const float*)d_in[6];
  const float* ei     = (const float*)d_in[7];
  const float* W1  = (const float*)d_in[8];
  const float* b1  = (const float*)d_in[9];
  const float* Wi1 = (const float*)d_in[10];
  const float* bi1 = (const float*)d_in[11];
  const float* W2  = (const float*)d_in[12];
  const float* b2  = (const float*)d_in[13];
  const float* Wi2 = (const float*)d_in[14];
  const float* bi2 = (const float*)d_in[15];

  int E     = in_sizes[3];   // 2 * NNZ edges
  int batch = in_sizes[0];

  size_t npd = (size_t)NNODES * DFEAT;
  float* e0 = (float*)d_ws;
  float* g1 = e0 + npd;
  float* g2 = g1 + npd;
  float* p1 = g2 + npd;
  float* p2 = p1 + npd;
  float* out = (float*)d_out;

  dim3 blk(256);
  int elemBlocks = (int)((npd + 255) / 256);
  int edgeBlocks = (E + 7) / 8;                      // 8 waves (edges) per block
  int rowTiles   = (NNODES + 15) / 16;
  int gemmBlocks = (rowTiles * 4 + 7) / 8;           // 8 tile-waves per block
  int predBlocks = (batch + 7) / 8;

  concat_embed_kernel<<<elemBlocks, blk, 0, stream>>>(eu, ei, e0);
  zero_loss_kernel<<<1, 32, 0, stream>>>(out + 2 * batch);

  // ---- layer 1 ----
  init_p_kernel<<<elemBlocks, blk, 0, stream>>>(e0, p1, p2);
  spmm_dual_kernel<<<edgeBlocks, blk, 0, stream>>>(rows, cols, vals, e0, p1, p2, E);
  gemm_bias_relu_kernel<<<gemmBlocks, blk, 0, stream>>>(p1, p2, W1, b1, Wi1, bi1,
                                                        g1, NNODES);
  // ---- layer 2 ----
  init_p_kernel<<<elemBlocks, blk, 0, stream>>>(g1, p1, p2);
  spmm_dual_kernel<<<edgeBlocks, blk, 0, stream>>>(rows, cols, vals, g1, p1, p2, E);
  gemm_bias_relu_kernel<<<gemmBlocks, blk, 0, stream>>>(p1, p2, W2, b2, Wi2, bi2,
                                                        g2, NNODES);

  pred_loss_kernel<<<predBlocks, blk, 0, stream>>>(user, item_i, item_j,
                                                   e0, g1, g2, out, batch);
}